// ViM_24283745092157
// MI455X (gfx1250) — compile-verified
//
#include <hip/hip_runtime.h>
#include <hip/hip_bf16.h>

// ---------------- model dims ----------------
#define DM    512
#define DI    1024
#define NST   16
#define KCONV 4
#define DRK   32
#define NLAY  8
#define NCLS  1000
#define BATCH 8
#define LSEQ  197
#define NPAT  196
#define PDIM  3840              // 16*16*15
#define ROWS  (BATCH * LSEQ)    // 1576
#define PROWS (BATCH * NPAT)    // 1568

typedef __attribute__((ext_vector_type(2))) float v2f;
typedef __attribute__((ext_vector_type(8))) float v8f;
typedef unsigned int u32x4 __attribute__((ext_vector_type(4)));
typedef int i32x8 __attribute__((ext_vector_type(8)));
typedef int i32x4 __attribute__((ext_vector_type(4)));

#if defined(__has_builtin)
#if __has_builtin(__builtin_amdgcn_tensor_load_to_lds) && \
    __has_builtin(__builtin_amdgcn_s_wait_tensorcnt)
#define HAVE_TDM 1
#endif
#endif

__device__ __forceinline__ float siluf(float x) { return x / (1.f + expf(-x)); }
__device__ __forceinline__ float softplusf(float x) {
  return fmaxf(x, 0.f) + log1pf(expf(-fabsf(x)));
}

// ---------------- block reduction (sum, sumsq) ----------------
__device__ __forceinline__ void block_reduce2(float& s, float& ss, float* rs, float* rss) {
  const int tid = threadIdx.x;
  rs[tid] = s; rss[tid] = ss;
  __syncthreads();
  for (int off = 128; off > 0; off >>= 1) {
    if (tid < off) { rs[tid] += rs[tid + off]; rss[tid] += rss[tid + off]; }
    __syncthreads();
  }
  s = rs[0]; ss = rss[0];
  __syncthreads();
}

#ifdef HAVE_TDM
// ---------------- TDM 2D tile load: global (row-major) -> LDS ----------------
// Loads tile_rows x tile_cols f32 tile starting at gsrc into LDS at lds_off.
// Remaining extents (rem_cols/rem_rows) give TDM OOB zero-fill on ragged edges.
// LDS padding: +4 DWORDs after every tile_cols DWORDs -> row pitch (tile_cols+4),
// which makes the ds_load A-fragment reads bank-conflict-free (pitch % 32 == 4).
// pad_interval code c means 2^(c+1) DWORDs; tile_cols must be 2^(c+1).
// 6-arg builtin on this toolchain: (g0, g1, g2, g3, g2b, cpol);
// groups 2/3 are unused for 2D tiles (tile_dim2 == 0) -> zero-filled.
__device__ __forceinline__ void tdm_load_2d(const float* gsrc, unsigned lds_off,
                                            int rem_cols, int rem_rows,
                                            int tile_cols, int tile_rows,
                                            int src_stride_elems, int pad_icode) {
  unsigned long long ga = (unsigned long long)(uintptr_t)gsrc;
  u32x4 g0;
  g0[0] = 1u;                                            // count=1, user mode
  g0[1] = lds_off;                                       // LDS byte address
  g0[2] = (unsigned)(ga & 0xffffffffu);                  // global_addr[31:0]
  g0[3] = (unsigned)((ga >> 32) & 0x01ffffffu) | 0x80000000u; // addr[56:32] | type=2
  const unsigned td0 = (unsigned)rem_cols;               // tensor extent dim0
  const unsigned td1 = (unsigned)rem_rows;               // tensor extent dim1
  const unsigned long long s0 = (unsigned long long)src_stride_elems;
  i32x8 g1;
  g1[0] = (int)((2u << 16)                 // data_size = 4 bytes
              | (1u << 20)                 // pad_enable
              | ((unsigned)pad_icode << 22)// pad_interval code
              | (3u << 25));               // pad_amount code 3 -> 4 DWORDs
  g1[1] = (int)((td0 & 0xffffu) << 16);
  g1[2] = (int)((td0 >> 16) | ((td1 & 0xffffu) << 16));
  g1[3] = (int)((td1 >> 16) | ((unsigned)tile_cols << 16));
  g1[4] = (int)((unsigned)tile_rows & 0xffffu);          // tile_dim1; tile_dim2=0
  g1[5] = (int)(unsigned)(s0 & 0xffffffffu);             // tensor_dim0_stride lo
  g1[6] = (int)(unsigned)((s0 >> 32) & 0xffffu);         // stride hi; dim1_stride=0
  g1[7] = 0;
  const i32x4 z4 = {0, 0, 0, 0};
  const i32x8 z8 = {0, 0, 0, 0, 0, 0, 0, 0};
  __builtin_amdgcn_tensor_load_to_lds(g0, g1, z4, z4, z8, 0);
}
#endif

// ---------------- f32 WMMA GEMM: C[M,N] = A[M,K] @ B[K,N] (+bias)(+act)(+=C) -------
// Block = 4 waves; block tile = 16 (M) x 64 (N), one 16x16 WMMA tile per wave.
// Wave 0 DMA-stages the 16 x KC activation tile into LDS via the Tensor Data Mover
// (zero-filled on the ragged M edge), waits TENSORcnt, then all 4 waves share it.
// B (weights, L2-resident) is read per-lane from global with a next-chunk prefetch.
// N must be a multiple of 64 and K a multiple of KC. KC is compile-time so the
// inner loop unrolls (multiple WMMAs in flight behind staggered load waits).
// WMMA fragments (V_WMMA_F32_16X16X4_F32, wave32):
//   A 16x4: lane m(0..15) holds K={0,1}; lanes 16..31 hold K={2,3}
//   B 4x16: lane n(0..15) holds K-rows {0,1}; lanes 16..31 hold {2,3}
//   C 16x16: vgpr r -> row r (lanes 0..15) / row r+8 (lanes 16..31)
template <int KC>
__global__ __launch_bounds__(128) void gemm_wmma_f32(
    const float* __restrict__ A, const float* __restrict__ B,
    const float* __restrict__ bias, float* __restrict__ C,
    int M, int N, int K, int lda, int ldb, int ldc, int act, int accum) {
  constexpr int PITCH = KC + 4;
  constexpr int ICODE = (KC == 128) ? 6 : (KC == 64) ? 5 : (KC == 32) ? 4 : 6;
  __shared__ float As[16 * PITCH];
  const int lane = threadIdx.x & 31;
  const int wv   = threadIdx.y;                 // 0..3
  const int tm   = blockIdx.y;
  const int tn   = blockIdx.x * 4 + wv;
  const int half = lane >> 4;
  const int l16  = lane & 15;
  const int col  = tn * 16 + l16;

#ifdef HAVE_TDM
  const unsigned lds_off = (unsigned)(uintptr_t)(&As[0]);
#endif

  v8f acc = {};
  for (int k0 = 0; k0 < K; k0 += KC) {
#ifdef HAVE_TDM
    if (wv == 0) {
      tdm_load_2d(A + (size_t)tm * 16 * lda + k0, lds_off,
                  K - k0, M - tm * 16, KC, 16, lda, ICODE);
      __builtin_amdgcn_s_wait_tensorcnt(0);
    }
#else
    for (int i = threadIdx.y * 32 + lane; i < 16 * KC; i += 128) {
      const int r = i / KC, c = i % KC;
      const int gr = tm * 16 + r;
      As[r * PITCH + c] = (gr < M) ? A[(size_t)gr * lda + k0 + c] : 0.f;
    }
#endif
    __syncthreads();

    if (k0 + KC < K)  // pull next weight chunk toward L2/L0
      __builtin_prefetch(B + (size_t)(k0 + KC) * ldb + col, 0, 1);

    const float* Arow = &As[l16 * PITCH + 2 * half];
    const float* Bp   = B + (size_t)(k0 + 2 * half) * ldb + col;
#pragma unroll 8
    for (int kk = 0; kk < KC; kk += 4) {
      v2f a, b;
      a.x = Arow[kk];
      a.y = Arow[kk + 1];
      b.x = Bp[(size_t)kk * ldb];
      b.y = Bp[(size_t)kk * ldb + ldb];
      acc = __builtin_amdgcn_wmma_f32_16x16x4_f32(false, a, false, b, (short)0, acc,
                                                  false, false);
    }
    __syncthreads();
  }

  const float bv = bias ? bias[col] : 0.f;
#pragma unroll
  for (int r = 0; r < 8; ++r) {
    const int crow = tm * 16 + half * 8 + r;
    if (crow < M) {
      float v = acc[r] + bv;
      if (act == 1) v = softplusf(v);
      const size_t idx = (size_t)crow * ldc + col;
      if (accum) C[idx] += v; else C[idx] = v;
    }
  }
}

// ---------------- patchify with 4 spatial shifts (15 channels) ----------------
__global__ __launch_bounds__(256) void patch_gather(const float* __restrict__ img,
                                                    float* __restrict__ Ap) {
  const long long idx = (long long)blockIdx.x * 256 + threadIdx.x;
  const long long total = (long long)PROWS * PDIM;
  if (idx >= total) return;
  const int row = (int)(idx / PDIM);
  const int j   = (int)(idx % PDIM);
  const int b = row / NPAT, p = row % NPAT, ph = p / 14, pw = p % 14;
  const int py = j / 240, r = j % 240, px = r / 15, c = r % 15;
  const int g = c / 3, cc = c % 3;
  const int dyv[5] = {0, 0, 0, 1, -1};
  const int dxv[5] = {0, 1, -1, 0, 0};
  const int y = ph * 16 + py - dyv[g];
  const int x = pw * 16 + px - dxv[g];
  float v = 0.f;
  if (y >= 0 && y < 224 && x >= 0 && x < 224)
    v = img[((size_t)b * 3 + cc) * (224 * 224) + (size_t)y * 224 + x];
  Ap[idx] = v;
}

// ---------------- row LayerNorm (in-place safe) ----------------
__global__ __launch_bounds__(256) void layernorm_rows(
    const float* __restrict__ X, const float* __restrict__ w,
    const float* __restrict__ b, float* __restrict__ Y, int D) {
  __shared__ float rs[256], rss[256];
  const int row = blockIdx.x;
  const float* xr = X + (size_t)row * D;
  float s = 0.f, ss = 0.f;
  for (int j = threadIdx.x; j < D; j += 256) { const float v = xr[j]; s += v; ss += v * v; }
  block_reduce2(s, ss, rs, rss);
  const float mean = s / D;
  const float var  = ss / D - mean * mean;
  const float rstd = rsqrtf(var + 1e-5f);
  float* yr = Y + (size_t)row * D;
  for (int j = threadIdx.x; j < D; j += 256)
    yr[j] = (xr[j] - mean) * rstd * w[j] + b[j];
}

// ---------------- assemble tokens: patches + cls token + pos_emb ----------------
__global__ __launch_bounds__(256) void assemble_x(const float* __restrict__ pe,
                                                  const float* __restrict__ cls,
                                                  const float* __restrict__ pos,
                                                  float* __restrict__ X) {
  const int idx = blockIdx.x * 256 + threadIdx.x;
  if (idx >= ROWS * DM) return;
  const int b = idx / (LSEQ * DM);
  const int t = (idx / DM) % LSEQ;
  const int d = idx % DM;
  const float v = (t < NPAT) ? pe[((size_t)b * NPAT + t) * DM + d] : cls[d];
  X[idx] = v + pos[(size_t)t * DM + d];
}

// ---------------- depthwise causal conv (K=4) + SiLU ----------------
__global__ __launch_bounds__(256) void conv_silu(const float* __restrict__ xz,
                                                 const float* __restrict__ cw,
                                                 const float* __restrict__ cb,
                                                 float* __restrict__ U) {
  const int idx = blockIdx.x * 256 + threadIdx.x;
  if (idx >= ROWS * DI) return;
  const int row = idx / DI;     // b*197 + t
  const int d   = idx % DI;
  const int b = row / LSEQ, t = row % LSEQ;
  float acc = cb[d];
#pragma unroll
  for (int k = 0; k < KCONV; ++k) {
    const int tt = t + k - (KCONV - 1);
    if (tt >= 0)
      acc += cw[d * KCONV + k] * xz[((size_t)b * LSEQ + tt) * (2 * DI) + d];
  }
  U[idx] = siluf(acc);
}

// ---------------- selective scan: serial over L, parallel over B*DI ----------------
__global__ __launch_bounds__(256) void mamba_scan(
    const float* __restrict__ dt, const float* __restrict__ U,
    const float* __restrict__ dbc, const float* __restrict__ A_log,
    const float* __restrict__ Dp, float* __restrict__ Ys) {
  const int gid = blockIdx.x * 256 + threadIdx.x;
  if (gid >= BATCH * DI) return;
  const int b = gid / DI, d = gid % DI;
  float A[NST], h[NST];
#pragma unroll
  for (int n = 0; n < NST; ++n) { A[n] = -expf(A_log[(size_t)d * NST + n]); h[n] = 0.f; }
  const float Dd = Dp[d];
  for (int t = 0; t < LSEQ; ++t) {
    const size_t row = (size_t)b * LSEQ + t;
    const float dtv = dt[row * DI + d];
    const float uv  = U[row * DI + d];
    const float du  = dtv * uv;
    const float* bc = dbc + row * (DRK + 2 * NST);
    float y = 0.f;
#pragma unroll
    for (int n = 0; n < NST; ++n) {
      const float dA = expf(dtv * A[n]);
      h[n] = dA * h[n] + du * bc[DRK + n];          // B part
      y += h[n] * bc[DRK + NST + n];                // C part
    }
    Ys[row * DI + d] = y + uv * Dd;
  }
}

// ---------------- gate: g = ys * silu(z) ----------------
__global__ __launch_bounds__(256) void gate_silu(const float* __restrict__ ys,
                                                 const float* __restrict__ xz,
                                                 float* __restrict__ G) {
  const int idx = blockIdx.x * 256 + threadIdx.x;
  if (idx >= ROWS * DI) return;
  const int row = idx / DI, d = idx % DI;
  const float z = xz[(size_t)row * (2 * DI) + DI + d];
  G[idx] = ys[idx] * siluf(z);
}

// ---------------- head: LN(last token) @ head_w + head_b ----------------
__global__ __launch_bounds__(256) void head_kernel(
    const float* __restrict__ X, const float* __restrict__ lw,
    const float* __restrict__ lb, const float* __restrict__ W,
    const float* __restrict__ hb, float* __restrict__ out) {
  __shared__ float rs[256], rss[256];
  __shared__ float xn[DM];
  const int b = blockIdx.x;
  const float* xr = X + ((size_t)b * LSEQ + (LSEQ - 1)) * DM;
  float s = 0.f, ss = 0.f;
  for (int j = threadIdx.x; j < DM; j += 256) { const float v = xr[j]; s += v; ss += v * v; }
  block_reduce2(s, ss, rs, rss);
  const float mean = s / DM;
  const float rstd = rsqrtf(ss / DM - mean * mean + 1e-5f);
  for (int j = threadIdx.x; j < DM; j += 256)
    xn[j] = (xr[j] - mean) * rstd * lw[j] + lb[j];
  __syncthreads();
  for (int c = threadIdx.x; c < NCLS; c += 256) {
    float a = hb[c];
    for (int d = 0; d < DM; ++d) a += xn[d] * W[(size_t)d * NCLS + c];
    out[(size_t)b * NCLS + c] = a;
  }
}

// ---------------- host launcher ----------------
static inline void launch_gemm(const float* A, const float* B, const float* bias, float* C,
                               int M, int N, int K, int lda, int ldb, int ldc,
                               int act, int accum, hipStream_t stream) {
  dim3 block(32, 4);
  dim3 grid(N / 64, (M + 15) / 16);   // N is always a multiple of 64 here
  if (K >= 128) {                     // K % 128 == 0 for all large GEMMs here
    gemm_wmma_f32<128><<<grid, block, 0, stream>>>(A, B, bias, C, M, N, K,
                                                   lda, ldb, ldc, act, accum);
  } else {                            // dt_proj: K == 32
    gemm_wmma_f32<32><<<grid, block, 0, stream>>>(A, B, bias, C, M, N, K,
                                                  lda, ldb, ldc, act, accum);
  }
}

extern "C" void kernel_launch(void* const* d_in, const int* in_sizes, int n_in,
                              void* d_out, int out_size, void* d_ws, size_t ws_size,
                              hipStream_t stream) {
  const float* img        = (const float*)d_in[0];
  const float* patch_ln_w = (const float*)d_in[1];
  const float* patch_ln_b = (const float*)d_in[2];
  const float* patch_w    = (const float*)d_in[3];
  const float* patch_b    = (const float*)d_in[4];
  const float* pos_emb    = (const float*)d_in[5];
  const float* cls_tok    = (const float*)d_in[6];
  const float* ln_w       = (const float*)d_in[7];
  const float* ln_b       = (const float*)d_in[8];
  const float* in_w       = (const float*)d_in[9];
  const float* conv_w     = (const float*)d_in[10];
  const float* conv_b     = (const float*)d_in[11];
  const float* xproj_w    = (const float*)d_in[12];
  const float* dt_w       = (const float*)d_in[13];
  const float* dt_b       = (const float*)d_in[14];
  const float* A_log      = (const float*)d_in[15];
  const float* Dp         = (const float*)d_in[16];
  const float* out_w      = (const float*)d_in[17];
  const float* head_ln_w  = (const float*)d_in[18];
  const float* head_ln_b  = (const float*)d_in[19];
  const float* head_w     = (const float*)d_in[20];
  const float* head_b     = (const float*)d_in[21];

  // workspace partition (floats)
  float* ws  = (float*)d_ws;
  float* ap  = ws;                         // 1568*3840
  float* pe  = ap  + (size_t)PROWS * PDIM; // 1568*512
  float* x   = pe  + (size_t)PROWS * DM;   // 1576*512
  float* y   = x   + (size_t)ROWS * DM;    // 1576*512
  float* xz  = y   + (size_t)ROWS * DM;    // 1576*2048
  float* u   = xz  + (size_t)ROWS * 2 * DI;// 1576*1024
  float* dbc = u   + (size_t)ROWS * DI;    // 1576*64
  float* dtb = dbc + (size_t)ROWS * (DRK + 2 * NST); // 1576*1024
  float* g   = dtb + (size_t)ROWS * DI;    // 1576*1024
  float* ysb = g   + (size_t)ROWS * DI;    // 1576*1024

  // 1. patchify (5-way shifted concat) -> Ap, then LayerNorm rows in-place
  {
    const long long total = (long long)PROWS * PDIM;
    patch_gather<<<(unsigned)((total + 255) / 256), 256, 0, stream>>>(img, ap);
    layernorm_rows<<<PROWS, 256, 0, stream>>>(ap, patch_ln_w, patch_ln_b, ap, PDIM);
  }
  // 2. patch embedding GEMM: pe = Ap @ patch_w + patch_b
  launch_gemm(ap, patch_w, patch_b, pe, PROWS, DM, PDIM, PDIM, DM, DM, 0, 0, stream);
  // 3. tokens: x = [pe, cls] + pos_emb
  assemble_x<<<(ROWS * DM + 255) / 256, 256, 0, stream>>>(pe, cls_tok, pos_emb, x);

  // 4. Mamba layers
  for (int i = 0; i < NLAY; ++i) {
    layernorm_rows<<<ROWS, 256, 0, stream>>>(x, ln_w + (size_t)i * DM,
                                             ln_b + (size_t)i * DM, y, DM);
    // in_proj: xz = y @ in_w[i]   (1576 x 2048)
    launch_gemm(y, in_w + (size_t)i * DM * 2 * DI, nullptr, xz,
                ROWS, 2 * DI, DM, DM, 2 * DI, 2 * DI, 0, 0, stream);
    // depthwise conv + silu -> u
    conv_silu<<<(ROWS * DI + 255) / 256, 256, 0, stream>>>(
        xz, conv_w + (size_t)i * DI * KCONV, conv_b + (size_t)i * DI, u);
    // x_proj: dbc = u @ xproj_w[i]   (1576 x 64)
    launch_gemm(u, xproj_w + (size_t)i * DI * (DRK + 2 * NST), nullptr, dbc,
                ROWS, DRK + 2 * NST, DI, DI, DRK + 2 * NST, DRK + 2 * NST, 0, 0, stream);
    // dt = softplus(dbc[:, :32] @ dt_w[i] + dt_b[i])   (1576 x 1024)
    launch_gemm(dbc, dt_w + (size_t)i * DRK * DI, dt_b + (size_t)i * DI, dtb,
                ROWS, DI, DRK, DRK + 2 * NST, DI, DI, 1, 0, stream);
    // selective scan
    mamba_scan<<<(BATCH * DI + 255) / 256, 256, 0, stream>>>(
        dtb, u, dbc, A_log + (size_t)i * DI * NST, Dp + (size_t)i * DI, ysb);
    // gate: g = ys * silu(z)
    gate_silu<<<(ROWS * DI + 255) / 256, 256, 0, stream>>>(ysb, xz, g);
    // out_proj + residual: x += g @ out_w[i]
    launch_gemm(g, out_w + (size_t)i * DI * DM, nullptr, x,
                ROWS, DM, DI, DI, DM, DM, 0, 1, stream);
  }

  // 5. head
  head_kernel<<<BATCH, 256, 0, stream>>>(x, head_ln_w, head_ln_b, head_w, head_b,
                                         (float*)d_out);
}